// SurfEval_16415365005744
// MI455X (gfx1250) — compile-verified
//
#include <hip/hip_runtime.h>
#include <float.h>

// Problem constants (match reference)
#define Bn    16
#define Mm    127      // last ctrl row index
#define Nn    127      // last ctrl col index
#define Pd    3
#define Qd    3
#define SUs   256
#define SVs   256
#define KLEN  132      // M + P + 2
#define CPD   128      // M+1 == N+1

typedef float v2f __attribute__((ext_vector_type(2)));
typedef float v8f __attribute__((ext_vector_type(8)));

__device__ __forceinline__ v8f wmma4(v2f a, v2f b, v8f c) {
    // D = A(16x4 f32) * B(4x16 f32) + C(16x16 f32)
    return __builtin_amdgcn_wmma_f32_16x16x4_f32(
        /*neg_a=*/false, a, /*neg_b=*/false, b,
        /*c_mod=*/(short)0, c, /*reuse_a=*/false, /*reuse_b=*/false);
}

__device__ __forceinline__ float sel4(int p, float n0, float n1, float n2, float n3) {
    float r = 0.0f;
    r = (p == 0) ? n0 : r;
    r = (p == 1) ? n1 : r;
    r = (p == 2) ? n2 : r;
    r = (p == 3) ? n3 : r;
    return r;
}

// ---------------------------------------------------------------------------
// Kernel 1: normalize knots, find spans, Cox-de Boor basis (tiny prep work).
// grid = (B, 2 axes), block = 256 (one thread per sample).
// ---------------------------------------------------------------------------
__global__ __launch_bounds__(SUs)
void basis_kernel(const float* __restrict__ knot_u, const float* __restrict__ knot_v,
                  const float* __restrict__ uarr,  const float* __restrict__ varr,
                  float* __restrict__ Nu, float* __restrict__ Nv,
                  int* __restrict__ uspan, int* __restrict__ vspan)
{
    const int b    = blockIdx.x;
    const int axis = blockIdx.y;
    const float* knots = (axis == 0) ? (knot_u + b * KLEN) : (knot_v + b * KLEN);
    const float* tarr  = (axis == 0) ? uarr : varr;
    float* Nout = (axis == 0) ? Nu : Nv;
    int*   Sout = (axis == 0) ? uspan : vspan;

    __shared__ float K[KLEN];
    if (threadIdx.x == 0) {
        // cumsum with negative-knot replacement, then normalize to [0,1]
        float acc = 0.0f;
        for (int i = 0; i < KLEN; ++i) {
            float kv = knots[i];
            acc += (kv < 0.0f) ? 1e-4f : kv;
            K[i] = acc;
        }
        const float k0  = K[0];
        const float inv = 1.0f / (K[KLEN - 1] - k0);
        for (int i = 0; i < KLEN; ++i) K[i] = (K[i] - k0) * inv;
    }
    __syncthreads();

    const int s   = threadIdx.x;
    const float t = tarr[s];

    // span: argmin over masked (t - K[3+j]) for j in [0,126)
    int   best  = 0;
    float bestv = FLT_MAX;
    for (int j = 0; j < KLEN - 2 * Pd; ++j) {
        float d = t - K[Pd + j];
        float m = (d > 1e-8f) ? d : 1.0f;
        if (m < bestv) { bestv = m; best = j; }
    }
    int span = best + Pd;
    span = (span < Pd) ? Pd : span;
    span = (span > Mm) ? Mm : span;
    Sout[b * SUs + s] = span;

    // Cox-de Boor (deg 3) with the reference's denom==0 quirk
    float Ni[Pd + 1];
    Ni[0] = 1.0f; Ni[1] = 0.0f; Ni[2] = 0.0f; Ni[3] = 0.0f;
    for (int k = 1; k <= Pd; ++k) {
        float saved = 0.0f;
        for (int r = 0; r < k; ++r) {
            float K1 = K[span + r + 1];
            float K2 = K[span + 1 - k + r];
            float denom = (K1 - t) + (t - K2);
            float temp  = (denom == 0.0f) ? 1e-4f : Ni[r] / denom;
            Ni[r] = saved + (K1 - t) * temp;
            saved = (t - K2) * temp;
        }
        Ni[k] = saved;
    }
    for (int p = 0; p <= Pd; ++p) Nout[(b * 4 + p) * SUs + s] = Ni[p];
}

// ---------------------------------------------------------------------------
// Kernel 2: one wave per 16x16 output tile. Two WMMA stages:
//   T = W_u * C_window   (per d-component)
//   out += T * W_v^T     (T through LDS for D-layout -> A-layout)
// ---------------------------------------------------------------------------
__global__ __launch_bounds__(32)
void eval_kernel(const float* __restrict__ ctrl,
                 const float* __restrict__ Nu, const float* __restrict__ Nv,
                 const int* __restrict__ uspan, const int* __restrict__ vspan,
                 float* __restrict__ out)
{
    const int lane = threadIdx.x;      // 0..31, full wave32
    const int l    = lane & 15;
    const int hi   = lane >> 4;        // 0 or 1
    const int b    = blockIdx.y;
    const int tu   = blockIdx.x >> 4;  // SV/16 == 16 tiles along v
    const int tv   = blockIdx.x & 15;
    const int u0   = tu * 16, v0 = tv * 16;

    // Wave-uniform window bounds (spans monotone in sample index)
    const int su_lo = uspan[b * SUs + u0];
    const int su_hi = uspan[b * SUs + u0 + 15];
    const int umin  = su_lo - Pd;
    const int nIc   = ((su_hi - su_lo + Pd + 1) + 15) >> 4;

    const int sv_lo = vspan[b * SVs + v0];
    const int sv_hi = vspan[b * SVs + v0 + 15];
    const int vmin  = sv_lo - Qd;
    const int nJc   = ((sv_hi - sv_lo + Qd + 1) + 15) >> 4;

    // Per-lane row (u = u0+l) and column (v = v0+l) basis data
    const int   su  = uspan[b * SUs + u0 + l];
    const float nu0 = Nu[(b * 4 + 0) * SUs + u0 + l];
    const float nu1 = Nu[(b * 4 + 1) * SUs + u0 + l];
    const float nu2 = Nu[(b * 4 + 2) * SUs + u0 + l];
    const float nu3 = Nu[(b * 4 + 3) * SUs + u0 + l];
    const int   sv  = vspan[b * SVs + v0 + l];
    const float nv0 = Nv[(b * 4 + 0) * SVs + v0 + l];
    const float nv1 = Nv[(b * 4 + 1) * SVs + v0 + l];
    const float nv2 = Nv[(b * 4 + 2) * SVs + v0 + l];
    const float nv3 = Nv[(b * 4 + 3) * SVs + v0 + l];

    __shared__ float Tl[3][16][17];    // stage-1 result, padded vs bank conflicts

    v8f acc0 = {}, acc1 = {}, acc2 = {};

    for (int jc = 0; jc < nJc; ++jc) {
        // ---- stage 1: T[u, jlocal, d] for this 16-column chunk ----
        v8f t0 = {}, t1 = {}, t2 = {};
        int c = vmin + jc * 16 + l;                    // this lane's column (B-frag N=l)
        c = (c < 0) ? 0 : ((c > Nn) ? Nn : c);

        for (int ic = 0; ic < nIc; ++ic) {
            #pragma unroll
            for (int kg = 0; kg < 4; ++kg) {
                const int r0 = umin + ic * 16 + kg * 4 + hi * 2;   // lane's K rows
                const int r1 = r0 + 1;
                // A fragment: sparse u-weight matrix W_u
                v2f A;
                A.x = sel4(r0 - (su - Pd), nu0, nu1, nu2, nu3);
                A.y = sel4(r1 - (su - Pd), nu0, nu1, nu2, nu3);
                // B fragments: control-point window (rows clamped; weights are 0 there)
                const int rc0 = (r0 < 0) ? 0 : ((r0 > Mm) ? Mm : r0);
                const int rc1 = (r1 < 0) ? 0 : ((r1 > Mm) ? Mm : r1);
                const float* c0 = ctrl + ((size_t)(b * CPD + rc0) * CPD + c) * 3;
                const float* c1 = ctrl + ((size_t)(b * CPD + rc1) * CPD + c) * 3;
                v2f B0 = { c0[0], c1[0] };
                v2f B1 = { c0[1], c1[1] };
                v2f B2 = { c0[2], c1[2] };
                t0 = wmma4(A, B0, t0);
                t1 = wmma4(A, B1, t1);
                t2 = wmma4(A, B2, t2);
            }
        }

        // D-layout -> LDS (VGPR g: lanes0-15 -> M=g, lanes16-31 -> M=g+8, N=lane%16)
        __syncthreads();
        #pragma unroll
        for (int g = 0; g < 8; ++g) {
            const int Mrow = g + 8 * hi;
            Tl[0][Mrow][l] = t0[g];
            Tl[1][Mrow][l] = t1[g];
            Tl[2][Mrow][l] = t2[g];
        }
        __syncthreads();

        // ---- stage 2: out += T * W_v^T ----
        #pragma unroll
        for (int kg = 0; kg < 4; ++kg) {
            const int kk = kg * 4 + hi * 2;            // lane's K rows (local j)
            const int j0 = vmin + jc * 16 + kk;
            v2f Bv;
            Bv.x = sel4(j0     - (sv - Qd), nv0, nv1, nv2, nv3);
            Bv.y = sel4(j0 + 1 - (sv - Qd), nv0, nv1, nv2, nv3);
            v2f A0 = { Tl[0][l][kk], Tl[0][l][kk + 1] };
            v2f A1 = { Tl[1][l][kk], Tl[1][l][kk + 1] };
            v2f A2 = { Tl[2][l][kk], Tl[2][l][kk + 1] };
            acc0 = wmma4(A0, Bv, acc0);
            acc1 = wmma4(A1, Bv, acc1);
            acc2 = wmma4(A2, Bv, acc2);
        }
    }

    // Store 16x16x3 tile
    #pragma unroll
    for (int g = 0; g < 8; ++g) {
        const int Mrow = g + 8 * hi;
        float* o = out + (((size_t)b * SUs + (u0 + Mrow)) * SVs + (v0 + l)) * 3;
        o[0] = acc0[g];
        o[1] = acc1[g];
        o[2] = acc2[g];
    }
}

// ---------------------------------------------------------------------------
extern "C" void kernel_launch(void* const* d_in, const int* in_sizes, int n_in,
                              void* d_out, int out_size, void* d_ws, size_t ws_size,
                              hipStream_t stream)
{
    const float* ctrl   = (const float*)d_in[0];   // (16,128,128,3) f32
    const float* knot_u = (const float*)d_in[1];   // (16,132) f32
    const float* knot_v = (const float*)d_in[2];   // (16,132) f32
    const float* uarr   = (const float*)d_in[3];   // (256,) f32
    const float* varr   = (const float*)d_in[4];   // (256,) f32
    float* out = (float*)d_out;                    // (16,256,256,3) f32

    // Workspace layout: Nu | Nv | uspan | vspan  (~160 KB)
    float* Nu    = (float*)d_ws;
    float* Nv    = Nu + Bn * 4 * SUs;
    int*   uspan = (int*)(Nv + Bn * 4 * SVs);
    int*   vspan = uspan + Bn * SUs;

    basis_kernel<<<dim3(Bn, 2), SUs, 0, stream>>>(knot_u, knot_v, uarr, varr,
                                                  Nu, Nv, uspan, vspan);
    eval_kernel<<<dim3((SUs / 16) * (SVs / 16), Bn), 32, 0, stream>>>(
        ctrl, Nu, Nv, uspan, vspan, out);
}